// LightGCN_16509854286404
// MI455X (gfx1250) — compile-verified
//
#include <hip/hip_runtime.h>

#define NUM_USERS  150000
#define NUM_ITEMS  140000
#define NUM_BRANDS 10000
#define N_NODES    (NUM_USERS + NUM_ITEMS + NUM_BRANDS)   // 300000
#define OUT_NODES  (NUM_USERS + NUM_ITEMS)                // 290000
#define DIM        64
#define N_EDGES    1200000

#define NODE_VEC4  (N_NODES * 16)     // float4 elements per full table (4.8M)
#define OUT_VEC4   (OUT_NODES * 16)   // float4 elements of d_out
#define USER_VEC4  (NUM_USERS * 16)

// ---------------------------------------------------------------------------
// k_init: b0 = ego (concat of the three tables), b1 = 0, out = ego[:OUT_NODES]
// ---------------------------------------------------------------------------
__global__ __launch_bounds__(256) void k_init(const float4* __restrict__ u,
                                              const float4* __restrict__ it,
                                              const float4* __restrict__ br,
                                              float4* __restrict__ b0,
                                              float4* __restrict__ b1,
                                              float4* __restrict__ out) {
    int t = blockIdx.x * 256 + threadIdx.x;
    if (t >= NODE_VEC4) return;
    float4 v;
    if (t < USER_VEC4)      v = u[t];
    else if (t < OUT_VEC4)  v = it[t - USER_VEC4];
    else                    v = br[t - OUT_VEC4];
    b0[t] = v;
    b1[t] = make_float4(0.f, 0.f, 0.f, 0.f);
    if (t < OUT_VEC4) out[t] = v;   // accumulator starts at ego (pure write)
}

// ---------------------------------------------------------------------------
// k_spmm: hn[dst] += h[src] * (val * coef)  for dst < dst_limit.
// 16 lanes per edge, float4 per lane: one coalesced 256B row gather per edge,
// 4 relaxed agent-scope f32 atomics -> native global_atomic_add_f32 in L2.
// Last layer scatters straight into d_out (dst_limit = OUT_NODES, coef = 1/4).
// ---------------------------------------------------------------------------
__global__ __launch_bounds__(256) void k_spmm(const float* __restrict__ h,
                                              float*       __restrict__ hn,
                                              const int*   __restrict__ src,
                                              const int*   __restrict__ dst,
                                              const float* __restrict__ val,
                                              float coef, int dst_limit) {
    unsigned t    = blockIdx.x * 256u + threadIdx.x;
    unsigned e    = t >> 4;
    if (e >= N_EDGES) return;
    unsigned lane = (t & 15u) << 2;   // dim offset: 0,4,...,60

    // stream the edge lists ahead of the hardware prefetcher (global_prefetch_b8)
    if ((t & 15u) == 0u && (e + 8192u) < N_EDGES) {
        __builtin_prefetch(src + e + 8192u, 0, 0);
        __builtin_prefetch(dst + e + 8192u, 0, 0);
        __builtin_prefetch(val + e + 8192u, 0, 0);
    }

    const int d = dst[e];
    if (d >= dst_limit) return;       // mandatory OOB guard for direct-to-out layer
    const int   s = src[e];
    const float v = val[e] * coef;

    const float4 m = *(const float4*)(h + (size_t)s * DIM + lane);
    float* o = hn + (size_t)d * DIM + lane;
    __hip_atomic_fetch_add(o + 0, m.x * v, __ATOMIC_RELAXED, __HIP_MEMORY_SCOPE_AGENT);
    __hip_atomic_fetch_add(o + 1, m.y * v, __ATOMIC_RELAXED, __HIP_MEMORY_SCOPE_AGENT);
    __hip_atomic_fetch_add(o + 2, m.z * v, __ATOMIC_RELAXED, __HIP_MEMORY_SCOPE_AGENT);
    __hip_atomic_fetch_add(o + 3, m.w * v, __ATOMIC_RELAXED, __HIP_MEMORY_SCOPE_AGENT);
}

// ---------------------------------------------------------------------------
// k_post: out[:OUT_VEC4] = (out + hn) * scale ;
// optionally re-zero the buffer that becomes the next scatter target.
// ---------------------------------------------------------------------------
__global__ __launch_bounds__(256) void k_post(const float4* __restrict__ hn,
                                              float4*       __restrict__ hz,
                                              float4*       __restrict__ out,
                                              float scale, int zero_en) {
    int t = blockIdx.x * 256 + threadIdx.x;
    if (t >= NODE_VEC4) return;
    if (zero_en) hz[t] = make_float4(0.f, 0.f, 0.f, 0.f);
    if (t < OUT_VEC4) {
        float4 a = out[t];
        float4 v = hn[t];
        a.x = (a.x + v.x) * scale;
        a.y = (a.y + v.y) * scale;
        a.z = (a.z + v.z) * scale;
        a.w = (a.w + v.w) * scale;
        out[t] = a;
    }
}

// ---------------------------------------------------------------------------
extern "C" void kernel_launch(void* const* d_in, const int* in_sizes, int n_in,
                              void* d_out, int out_size, void* d_ws, size_t ws_size,
                              hipStream_t stream) {
    const float* user  = (const float*)d_in[0];
    const float* item  = (const float*)d_in[1];
    const float* brand = (const float*)d_in[2];
    const float* eval  = (const float*)d_in[3];
    const int*   esrc  = (const int*)d_in[4];
    const int*   edst  = (const int*)d_in[5];
    float* out = (float*)d_out;

    float* b0 = (float*)d_ws;                       // 76.8 MB
    float* b1 = b0 + (size_t)N_NODES * DIM;         // 76.8 MB

    const int gNode = (NODE_VEC4 + 255) / 256;               // 18750 blocks
    const int gOut  = (OUT_VEC4  + 255) / 256;               // 18125 blocks
    const int gEdge = (int)((N_EDGES * 16u + 255u) / 256u);  // 75000 blocks

    k_init<<<gNode, 256, 0, stream>>>((const float4*)user, (const float4*)item,
                                      (const float4*)brand,
                                      (float4*)b0, (float4*)b1, (float4*)out);

    // layer 1: h1 = A*ego   (b0 -> b1) ; acc += h1 ; re-zero b0 for layer 2
    k_spmm<<<gEdge, 256, 0, stream>>>(b0, b1, esrc, edst, eval, 1.0f, N_NODES);
    k_post<<<gNode, 256, 0, stream>>>((const float4*)b1, (float4*)b0,
                                      (float4*)out, 1.0f, 1);

    // layer 2: h2 = A*h1    (b1 -> b0) ; acc = (acc + h2) * 1/4 ; no re-zero
    k_spmm<<<gEdge, 256, 0, stream>>>(b1, b0, esrc, edst, eval, 1.0f, N_NODES);
    k_post<<<gOut, 256, 0, stream>>>((const float4*)b0, (float4*)b1,
                                     (float4*)out, 0.25f, 0);

    // layer 3: acc += (A*h2) * 1/4 scattered DIRECTLY into d_out
    // (rows >= OUT_NODES have no storage in d_out -> guarded in-kernel)
    k_spmm<<<gEdge, 256, 0, stream>>>(b0, out, esrc, edst, eval, 0.25f, OUT_NODES);
}